// Net_3633542332684
// MI455X (gfx1250) — compile-verified
//
#include <hip/hip_runtime.h>

typedef float v2f __attribute__((ext_vector_type(2)));
typedef float v8f __attribute__((ext_vector_type(8)));

// ---------------- zero scratch ----------------
__global__ void k_zero(float* __restrict__ p, int n) {
    int i = blockIdx.x * blockDim.x + threadIdx.x;
    if (i < n) p[i] = 0.0f;
}

// ---------------- A = x @ [W1_l | W1_r]  via V_WMMA_F32_16X16X4_F32 ----------------
// x: [n,128] row-major; w1l/w1r: [128,32] row-major; aout: [n,64] row-major.
// 256 threads = 8 waves; wave w computes 16 rows x 64 cols.
// W staged in LDS column-major with stride 132 (conflict-free b64 fragment loads).
#define WSTRIDE 132
__global__ __launch_bounds__(256) void k_gemm(const float* __restrict__ x,
                                              const float* __restrict__ w1l,
                                              const float* __restrict__ w1r,
                                              float* __restrict__ aout,
                                              int ntiles, int n_nodes) {
    __shared__ float wsh[64 * WSTRIDE];      // ~33 KB: W^T, wsh[col*132 + k]
    const int tid = threadIdx.x;
    for (int i = tid; i < 64 * 128; i += 256) {
        int col = i >> 7, k = i & 127;       // consecutive threads -> consecutive k: no bank conflicts
        wsh[col * WSTRIDE + k] = (col < 32) ? w1l[k * 32 + col] : w1r[k * 32 + (col - 32)];
    }
    __syncthreads();

    const int wave = tid >> 5, lane = tid & 31;
    const int hl = lane >> 4, lm = lane & 15;   // lane half / lane-in-half

    int tile = blockIdx.x * 8 + wave;
    if (tile >= ntiles) tile = ntiles - 1;      // clamp (no branch) -> EXEC stays all-1s for WMMA
    int row0 = tile * 16;
    if (row0 > n_nodes - 16) row0 = n_nodes - 16;

    const float* xr = x + (size_t)(row0 + lm) * 128;
    // per-column-tile base pointers: column (c*16 + lm) of W^T
    const float* wb0 = wsh + (size_t)(0 * 16 + lm) * WSTRIDE;
    const float* wb1 = wsh + (size_t)(1 * 16 + lm) * WSTRIDE;
    const float* wb2 = wsh + (size_t)(2 * 16 + lm) * WSTRIDE;
    const float* wb3 = wsh + (size_t)(3 * 16 + lm) * WSTRIDE;

    v8f acc[4] = {v8f{}, v8f{}, v8f{}, v8f{}};

    #pragma unroll 4
    for (int kt = 0; kt < 32; ++kt) {
        const int kk = kt * 4 + hl * 2;         // A layout: lanes 0-15 hold K0,K1; 16-31 hold K2,K3
        v2f a = *(const v2f*)(xr + kk);         // single global b64
        v2f b0 = *(const v2f*)(wb0 + kk);       // single ds_load_b64 each, conflict-free
        v2f b1f = *(const v2f*)(wb1 + kk);
        v2f b2f = *(const v2f*)(wb2 + kk);
        v2f b3f = *(const v2f*)(wb3 + kk);
        acc[0] = __builtin_amdgcn_wmma_f32_16x16x4_f32(false, a, false, b0,  (short)0, acc[0], false, false);
        acc[1] = __builtin_amdgcn_wmma_f32_16x16x4_f32(false, a, false, b1f, (short)0, acc[1], false, false);
        acc[2] = __builtin_amdgcn_wmma_f32_16x16x4_f32(false, a, false, b2f, (short)0, acc[2], false, false);
        acc[3] = __builtin_amdgcn_wmma_f32_16x16x4_f32(false, a, false, b3f, (short)0, acc[3], false, false);
    }

    // C/D layout: VGPR i, lanes 0-15 -> M=i ; lanes 16-31 -> M=8+i ; col = lane%16
    #pragma unroll
    for (int c = 0; c < 4; ++c) {
        #pragma unroll
        for (int i = 0; i < 8; ++i) {
            aout[(size_t)(row0 + hl * 8 + i) * 64 + c * 16 + lm] = acc[c][i];
        }
    }
}

// ---------------- layer-1 scatter: one wave per edge, 32 feature lanes ----------------
__global__ void k_scatter1(const int* __restrict__ ei, const float* __restrict__ aout,
                           float* __restrict__ s1, float* __restrict__ deg, int n_edges) {
    long long t = (long long)blockIdx.x * blockDim.x + threadIdx.x;
    int e = (int)(t >> 5);
    int j = (int)(t & 31);
    if (e >= n_edges) return;
    int src = ei[e];
    int dst = ei[n_edges + e];
    atomicAdd(&s1[(size_t)dst * 32 + j], aout[(size_t)src * 64 + j]);  // cols 0..31 = x@W1_l
    if (j == 0) atomicAdd(&deg[dst], 1.0f);
}

// ---------------- fused: h = relu(mean + self + b1); a2 = h@W2_l; r2 = h@W2_r + b2 ----------------
__global__ void k_combine(const float* __restrict__ s1, const float* __restrict__ aout,
                          const float* __restrict__ deg, const float* __restrict__ b1,
                          const float* __restrict__ w2l, const float* __restrict__ w2r,
                          const float* __restrict__ b2, float* __restrict__ a2,
                          float* __restrict__ r2, int n_nodes) {
    long long t = (long long)blockIdx.x * blockDim.x + threadIdx.x;
    int node = (int)(t >> 5);
    int j = (int)(t & 31);
    if (node >= n_nodes) return;
    float inv = 1.0f / fmaxf(deg[node], 1.0f);
    float h = s1[(size_t)node * 32 + j] * inv + aout[(size_t)node * 64 + 32 + j] + b1[j];
    h = fmaxf(h, 0.0f);
    float av = h * w2l[j];
    float rv = h * w2r[j];
    #pragma unroll
    for (int off = 16; off > 0; off >>= 1) {    // wave32 tree reduction
        av += __shfl_xor(av, off, 32);
        rv += __shfl_xor(rv, off, 32);
    }
    if (j == 0) {
        a2[node] = av;
        r2[node] = rv + b2[0];
    }
}

// ---------------- layer-2 scalar scatter ----------------
__global__ void k_scatter2(const int* __restrict__ ei, const float* __restrict__ a2,
                           float* __restrict__ s2, int n_edges) {
    int e = blockIdx.x * blockDim.x + threadIdx.x;
    if (e >= n_edges) return;
    int src = ei[e];
    int dst = ei[n_edges + e];
    atomicAdd(&s2[dst], a2[src]);
}

// ---------------- final combine ----------------
__global__ void k_final(const float* __restrict__ s2, const float* __restrict__ deg,
                        const float* __restrict__ r2, float* __restrict__ out, int n_nodes) {
    int n = blockIdx.x * blockDim.x + threadIdx.x;
    if (n >= n_nodes) return;
    out[n] = s2[n] / fmaxf(deg[n], 1.0f) + r2[n];
}

extern "C" void kernel_launch(void* const* d_in, const int* in_sizes, int n_in,
                              void* d_out, int out_size, void* d_ws, size_t ws_size,
                              hipStream_t stream) {
    const float* x   = (const float*)d_in[0];
    const int*   ei  = (const int*)d_in[1];
    const float* w1l = (const float*)d_in[2];
    const float* w1r = (const float*)d_in[3];
    const float* b1  = (const float*)d_in[4];
    const float* w2l = (const float*)d_in[5];
    const float* w2r = (const float*)d_in[6];
    const float* b2  = (const float*)d_in[7];
    float* out = (float*)d_out;

    const int n_nodes = in_sizes[0] / 128;   // 100000
    const int n_edges = in_sizes[1] / 2;     // 640000

    // workspace carve (floats): A[n*64] | S1[n*32] | deg[n] | s2[n] | a2[n] | r2[n]  (~40 MB)
    float* A   = (float*)d_ws;
    float* S1  = A   + (size_t)n_nodes * 64;
    float* deg = S1  + (size_t)n_nodes * 32;
    float* s2  = deg + n_nodes;
    float* a2  = s2  + n_nodes;
    float* r2  = a2  + n_nodes;

    // zero the accumulation targets (S1, deg, s2 are contiguous: 34*n floats)
    int zn = n_nodes * 34;
    k_zero<<<(zn + 255) / 256, 256, 0, stream>>>(S1, zn);

    int ntiles = (n_nodes + 15) / 16;        // 6250
    k_gemm<<<(ntiles + 7) / 8, 256, 0, stream>>>(x, w1l, w1r, A, ntiles, n_nodes);

    long long t1 = (long long)n_edges * 32;
    k_scatter1<<<(unsigned)((t1 + 255) / 256), 256, 0, stream>>>(ei, A, S1, deg, n_edges);

    long long t2 = (long long)n_nodes * 32;
    k_combine<<<(unsigned)((t2 + 255) / 256), 256, 0, stream>>>(S1, A, deg, b1, w2l, w2r, b2,
                                                                a2, r2, n_nodes);

    k_scatter2<<<(n_edges + 255) / 256, 256, 0, stream>>>(ei, a2, s2, n_edges);

    k_final<<<(n_nodes + 255) / 256, 256, 0, stream>>>(s2, deg, r2, out, n_nodes);
}